// GMMModel_89541478187028
// MI455X (gfx1250) — compile-verified
//
#include <hip/hip_runtime.h>
#include <hip/hip_bf16.h>
#include <math.h>

// ---- problem constants (from reference) ----
constexpr int cBS   = 2;
constexpr int cN    = 15135;
constexpr int cE    = 242160;
constexpr int cFIN  = 8;
constexpr int cH    = 64;
constexpr int cL    = 4;
constexpr int cK    = 2;
constexpr int cD    = 2;
constexpr int cHFC  = 256;
constexpr int cNC   = 10;
constexpr float cEPS = 1e-15f;

typedef __attribute__((ext_vector_type(2))) float v2f;
typedef __attribute__((ext_vector_type(8))) float v8f;

#if defined(__AMDGCN__) && __has_builtin(__builtin_amdgcn_wmma_f32_16x16x4_f32)
#define HAS_WMMA_F32X4 1
#else
#define HAS_WMMA_F32X4 0
#endif

__device__ __forceinline__ float eluf(float v) {
    return v > 0.0f ? v : expm1f(v);
}

// ---------------- WMMA GEMM: C[M x (NT*16)] = A[M x Kd] @ B[Kd x (NT*16)] (+bias) ----
// One wave (32 threads) per 16-row tile; NT column tiles per wave (A frag reused).
template<int NT>
__global__ void wmma_gemm_kernel(const float* __restrict__ A,
                                 const float* __restrict__ B,
                                 const float* __restrict__ bias,
                                 float* __restrict__ C,
                                 int M, int Kd) {
    constexpr int Nd = NT * 16;
    const int lane  = threadIdx.x;       // 0..31 (wave32)
    const int mbase = blockIdx.x * 16;
    const int li = lane & 15;
    const int hi = lane >> 4;            // 0 or 1
#if HAS_WMMA_F32X4
    const int arow = min(mbase + li, M - 1);   // clamp boundary tile reads
    const int kofs = hi * 2;                   // K-pair owned by this half-wave
    v8f acc[NT];
#pragma unroll
    for (int t = 0; t < NT; ++t)
        acc[t] = (v8f){0.f, 0.f, 0.f, 0.f, 0.f, 0.f, 0.f, 0.f};
    for (int kk = 0; kk < Kd; kk += 4) {
        v2f a;
        a.x = A[(size_t)arow * Kd + kk + kofs + 0];
        a.y = A[(size_t)arow * Kd + kk + kofs + 1];
#pragma unroll
        for (int t = 0; t < NT; ++t) {
            const int col = t * 16 + li;
            v2f b;
            b.x = B[(size_t)(kk + kofs + 0) * Nd + col];
            b.y = B[(size_t)(kk + kofs + 1) * Nd + col];
            acc[t] = __builtin_amdgcn_wmma_f32_16x16x4_f32(
                false, a, false, b, (short)0, acc[t], false, false);
        }
    }
#pragma unroll
    for (int t = 0; t < NT; ++t) {
        const int col = t * 16 + li;
        const float bv = bias ? bias[col] : 0.0f;
#pragma unroll
        for (int r = 0; r < 8; ++r) {
            const int row = mbase + r + hi * 8;    // C/D VGPR layout
            if (row < M) C[(size_t)row * Nd + col] = acc[t][r] + bv;
        }
    }
#else
    // scalar fallback (same output mapping)
    for (int t = 0; t < NT; ++t) {
        const int col = t * 16 + li;
        const float bv = bias ? bias[col] : 0.0f;
        for (int r = 0; r < 8; ++r) {
            const int row = mbase + r + hi * 8;
            if (row >= M) continue;
            float s = 0.f;
            for (int k = 0; k < Kd; ++k)
                s += A[(size_t)row * Kd + k] * B[(size_t)k * Nd + col];
            C[(size_t)row * Nd + col] = s + bv;
        }
    }
#endif
}

// ---------------- degree (in-degree, clamped, inverted) ----------------
__global__ void deg_kernel(const int* __restrict__ dst, float* __restrict__ deg) {
    int e = blockIdx.x * blockDim.x + threadIdx.x;
    if (e < cE) atomicAdd(&deg[dst[e]], 1.0f);
}

__global__ void invdeg_kernel(float* __restrict__ deg) {
    int n = blockIdx.x * blockDim.x + threadIdx.x;
    if (n < cN) deg[n] = 1.0f / fmaxf(deg[n], 1.0f);
}

// ---------------- Gaussian mixture weights per edge ----------------
__global__ void gauss_kernel(const float* __restrict__ pseudo,
                             const float* __restrict__ mu,
                             const float* __restrict__ sigma,
                             float* __restrict__ gauss) {
    int e = blockIdx.x * blockDim.x + threadIdx.x;
    if (e >= cE) return;
    const float p0 = pseudo[e * 2 + 0];
    const float p1 = pseudo[e * 2 + 1];
#pragma unroll
    for (int k = 0; k < cK; ++k) {
        const float d0 = p0 - mu[k * 2 + 0];
        const float d1 = p1 - mu[k * 2 + 1];
        const float s0 = sigma[k * 2 + 0];
        const float s1 = sigma[k * 2 + 1];
        const float a = -0.5f * (d0 * d0 / (cEPS + s0 * s0) +
                                 d1 * d1 / (cEPS + s1 * s1));
        gauss[e * 2 + k] = expf(a);
    }
}

// ---------------- edge gather + weighted scatter (L2-resident) ----------------
__global__ void edge_scatter_kernel(const int* __restrict__ src,
                                    const int* __restrict__ dst,
                                    const float* __restrict__ gauss,
                                    const float* __restrict__ hk,   // (BS,N,2H)
                                    float* __restrict__ acc) {      // (BS,N,H)
    const int tid = blockIdx.x * blockDim.x + threadIdx.x;
    if (tid >= cE * cH) return;
    const int e = tid >> 6;
    const int o = tid & (cH - 1);
    const int s = src[e];
    const int d = dst[e];
    const float g0 = gauss[e * 2 + 0];
    const float g1 = gauss[e * 2 + 1];
#pragma unroll
    for (int b = 0; b < cBS; ++b) {
        const size_t hb = ((size_t)b * cN + s) * (2 * cH);
        const float v = g0 * hk[hb + o] + g1 * hk[hb + cH + o];
        atomicAdd(&acc[((size_t)b * cN + d) * cH + o], v);
    }
}

// ---------------- finalize layer: mean + root + ELU; write h and xc slot ----------
__global__ void finalize_kernel(const float* __restrict__ acc,
                                const float* __restrict__ invdeg,
                                const float* __restrict__ hroot,   // includes bias
                                float* __restrict__ hout,          // (BS,N,H)
                                float* __restrict__ xcb,           // (BS,N,H,L)
                                int layer) {
    const int tid = blockIdx.x * blockDim.x + threadIdx.x;
    if (tid >= cBS * cN * cH) return;
    const int o  = tid & (cH - 1);
    const int bn = tid >> 6;
    const int n  = bn % cN;
    const float v = acc[tid] * invdeg[n] + hroot[tid];
    const float h = eluf(v);
    hout[tid] = h;
    xcb[(size_t)bn * (cH * cL) + o * cL + layer] = h;   // stack(xs,-1) layout
}

// ---------------- node embedding: xc @ fc_w + fc_b (wave per node) ----------------
__global__ void fc_node_kernel(const float* __restrict__ xcb,
                               const float* __restrict__ fcw,
                               const float* __restrict__ fcb,
                               float* __restrict__ node) {
    const int bn = blockIdx.x;
    const int lane = threadIdx.x;
    float s = 0.f;
    for (int i = lane; i < cH * cL; i += 32)
        s += xcb[(size_t)bn * (cH * cL) + i] * fcw[i];
#pragma unroll
    for (int off = 16; off > 0; off >>= 1) s += __shfl_down(s, off, 32);
    if (lane == 0) node[bn] = s + fcb[0];
}

// ---------------- lin1 partial: node(BS,N) @ lin1_w(N,HFC) ----------------
__global__ void lin1_partial_kernel(const float* __restrict__ node,
                                    const float* __restrict__ w,
                                    float* __restrict__ h1acc) {
    const int j = threadIdx.x;            // 0..HFC-1, coalesced over w rows
    const int b = blockIdx.y;
    const int chunk = (cN + gridDim.x - 1) / gridDim.x;
    const int n0 = blockIdx.x * chunk;
    const int n1 = min(n0 + chunk, cN);
    float s = 0.f;
    for (int n = n0; n < n1; ++n)
        s += node[b * cN + n] * w[(size_t)n * cHFC + j];
    atomicAdd(&h1acc[b * cHFC + j], s);
}

__global__ void h1_elu_kernel(const float* __restrict__ a,
                              const float* __restrict__ bias,
                              float* __restrict__ h1) {
    const int t = blockIdx.x * blockDim.x + threadIdx.x;
    if (t >= cBS * cHFC) return;
    h1[t] = eluf(a[t] + bias[t & (cHFC - 1)]);
}

// ---------------- lin2 + log_softmax (tiny) ----------------
__global__ void head_kernel(const float* __restrict__ h1,
                            const float* __restrict__ w2,
                            const float* __restrict__ b2,
                            float* __restrict__ out) {
    const int b = threadIdx.x;
    if (b >= cBS) return;
    float lg[cNC];
    float mx = -INFINITY;
    for (int c = 0; c < cNC; ++c) {
        float s = b2[c];
        for (int j = 0; j < cHFC; ++j) s += h1[b * cHFC + j] * w2[j * cNC + c];
        lg[c] = s;
        mx = fmaxf(mx, s);
    }
    float se = 0.f;
    for (int c = 0; c < cNC; ++c) se += expf(lg[c] - mx);
    const float lse = logf(se) + mx;
    for (int c = 0; c < cNC; ++c) out[b * cNC + c] = lg[c] - lse;
}

// ============================= host side =============================
extern "C" void kernel_launch(void* const* d_in, const int* in_sizes, int n_in,
                              void* d_out, int out_size, void* d_ws, size_t ws_size,
                              hipStream_t stream) {
    (void)in_sizes; (void)n_in; (void)out_size; (void)ws_size;

    const float* x      = (const float*)d_in[0];
    const int*   eidx   = (const int*)d_in[2];
    const int*   src    = eidx;
    const int*   dst    = eidx + cE;
    const float* pseudo = (const float*)d_in[3];
    const float* g1     = (const float*)d_in[4];
    const float* mu1    = (const float*)d_in[5];
    const float* sigma1 = (const float*)d_in[6];
    const float* root1  = (const float*)d_in[7];
    const float* b1     = (const float*)d_in[8];
    const float* gs     = (const float*)d_in[9];
    const float* mus    = (const float*)d_in[10];
    const float* sigmas = (const float*)d_in[11];
    const float* roots  = (const float*)d_in[12];
    const float* bsp    = (const float*)d_in[13];
    const float* fcw    = (const float*)d_in[14];
    const float* fcb    = (const float*)d_in[15];
    const float* l1w    = (const float*)d_in[16];
    const float* l1b    = (const float*)d_in[17];
    const float* l2w    = (const float*)d_in[18];
    const float* l2b    = (const float*)d_in[19];

    char* ws = (char*)d_ws;
    size_t off = 0;
    auto take = [&](size_t bytes) -> float* {
        float* p = (float*)(ws + off);
        off += (bytes + 255) & ~(size_t)255;
        return p;
    };
    const int M = cBS * cN;
    float* deg    = take((size_t)cN * 4);                 // -> invdeg after invert
    float* gaussb = take((size_t)cE * cK * 4);
    float* hk     = take((size_t)M * 2 * cH * 4);
    float* hroot  = take((size_t)M * cH * 4);
    float* accb   = take((size_t)M * cH * 4);
    float* hcur   = take((size_t)M * cH * 4);
    float* xcb    = take((size_t)M * cH * cL * 4);
    float* node   = take((size_t)M * 4);
    float* h1acc  = take((size_t)cBS * cHFC * 4);
    float* h1     = take((size_t)cBS * cHFC * 4);

    // degree once (shared by all layers/batches)
    hipMemsetAsync(deg, 0, (size_t)cN * 4, stream);
    deg_kernel<<<(cE + 255) / 256, 256, 0, stream>>>(dst, deg);
    invdeg_kernel<<<(cN + 255) / 256, 256, 0, stream>>>(deg);

    const int Mtiles = (M + 15) / 16;
    const float* hin = x;
    int Kd = cFIN;
    for (int l = 0; l < cL; ++l) {
        const float* g  = (l == 0) ? g1     : gs     + (size_t)(l - 1) * cH * cK * cH;
        const float* mu = (l == 0) ? mu1    : mus    + (size_t)(l - 1) * cK * cD;
        const float* sg = (l == 0) ? sigma1 : sigmas + (size_t)(l - 1) * cK * cD;
        const float* rt = (l == 0) ? root1  : roots  + (size_t)(l - 1) * cH * cH;
        const float* bb = (l == 0) ? b1     : bsp    + (size_t)(l - 1) * cH;

        gauss_kernel<<<(cE + 255) / 256, 256, 0, stream>>>(pseudo, mu, sg, gaussb);
        // hk = h @ g  (M x 128), exact fp32 via V_WMMA_F32_16X16X4_F32
        wmma_gemm_kernel<8><<<Mtiles, 32, 0, stream>>>(hin, g, nullptr, hk, M, Kd);
        // hroot = h @ root + bias  (M x 64)
        wmma_gemm_kernel<4><<<Mtiles, 32, 0, stream>>>(hin, rt, bb, hroot, M, Kd);

        hipMemsetAsync(accb, 0, (size_t)M * cH * 4, stream);
        edge_scatter_kernel<<<(cE * cH + 255) / 256, 256, 0, stream>>>(
            src, dst, gaussb, hk, accb);
        finalize_kernel<<<(M * cH + 255) / 256, 256, 0, stream>>>(
            accb, deg, hroot, hcur, xcb, l);
        hin = hcur;
        Kd = cH;
    }

    fc_node_kernel<<<M, 32, 0, stream>>>(xcb, fcw, fcb, node);
    hipMemsetAsync(h1acc, 0, (size_t)cBS * cHFC * 4, stream);
    dim3 lg(64, cBS, 1);
    lin1_partial_kernel<<<lg, cHFC, 0, stream>>>(node, l1w, h1acc);
    h1_elu_kernel<<<(cBS * cHFC + 255) / 256, 256, 0, stream>>>(h1acc, l1b, h1);
    head_kernel<<<1, 32, 0, stream>>>(h1, l2w, l2b, (float*)d_out);
}